// GridAttention_63909113364991
// MI455X (gfx1250) — compile-verified
//
#include <hip/hip_runtime.h>

// ---------------------------------------------------------------------------
// GridAttention fused kernel for gfx1250 (MI455X), fp32 end-to-end using
// V_WMMA_F32_16X16X4_F32 + Tensor Data Mover staging. Algebraically factored
// attention (single query token per cell) removes the full K/V projections:
//   scores[h,l] = U_h . x_l + qp_h.bk_h,   U_h = Wk_h^T qp_h
//   ctx_h       = Wv_h . (sum_l attn[h,l] x_l) + bv_h
// ---------------------------------------------------------------------------

typedef float v2f __attribute__((ext_vector_type(2)));
typedef float v8f __attribute__((ext_vector_type(8)));
typedef unsigned int u32x4 __attribute__((ext_vector_type(4)));
typedef int i32x8 __attribute__((ext_vector_type(8)));
typedef int i32x4 __attribute__((ext_vector_type(4)));

#define WMMA_F32(a, b, c) \
  __builtin_amdgcn_wmma_f32_16x16x4_f32(false, (a), false, (b), (short)0, (c), false, false)

#if __has_builtin(__builtin_amdgcn_tensor_load_to_lds) && \
    __has_builtin(__builtin_amdgcn_s_wait_tensorcnt)
#define USE_TDM 1
#else
#define USE_TDM 0
#endif

constexpr int DM = 256;     // model dim
constexpr int CELLS = 8;    // cells per workgroup (gy-consecutive)
constexpr int NTHREADS = 256;

// LDS carve-out (float offsets); padded strides 257 / 65 keep the strided
// WMMA A-fragment loads bank-conflict free (64 banks x 4B).
constexpr int OFF_Q  = 0;                   // Q   [16][257]  pooled means (rows 8..15 zero)
constexpr int OFF_QP = OFF_Q  + 16 * 257;   // QP  [16][257]  q-projection; reused as CTX
constexpr int OFF_U  = OFF_QP + 16 * 257;   // U   [2][8][257] per-pair key-side vectors
constexpr int OFF_CB = OFF_U  + 2 * 8 * 257;// CB  [8][8]     qp_h . bk_h
constexpr int OFF_XT = OFF_CB + 64;         // XT  [2][64][65] staged X^T chunk (64 dims)
constexpr int OFF_AT = OFF_XT + 2 * 64 * 65;// AT  [2][8][65] scores -> attn
constexpr int OFF_S2 = OFF_AT + 2 * 8 * 65; // S2  [8][8][257] attn-weighted token sums
constexpr int LDS_FLOATS = OFF_S2 + 8 * 8 * 257;   // 38208 floats = 152832 B

#if USE_TDM
// One TDM descriptor moves a whole per-cell chunk: 3D tile X=8 contiguous
// floats (iy), Y=8 rows stride 256 (ix), Z=64 planes stride 65536 (j);
// LDS destination padded +1 DWORD per 64 DWORDs -> XT row stride 65.
// 6-arg builtin (clang-23 flavor): (g0 v4u, g1 v8i, g2 v4i, g3 v4i, v8i, cpol)
__device__ __forceinline__ void tdm_load_cell_chunk(const float* gsrc, float* ldst) {
  unsigned ga_lo = (unsigned)__builtin_amdgcn_readfirstlane((int)(unsigned)(unsigned long long)(uintptr_t)gsrc);
  unsigned ga_hi = (unsigned)__builtin_amdgcn_readfirstlane((int)(unsigned)(((unsigned long long)(uintptr_t)gsrc) >> 32));
  unsigned la    = (unsigned)__builtin_amdgcn_readfirstlane((int)(unsigned)(uintptr_t)ldst); // LDS flat addr, low 32 = offset

  u32x4 g0;
  g0[0] = 1u;                                        // count=1, user descriptor
  g0[1] = la;                                        // lds_addr (bytes)
  g0[2] = ga_lo;                                     // global_addr[31:0]
  g0[3] = (ga_hi & 0x01FFFFFFu) | (2u << 30);        // global_addr[56:32] | type=2 (image)

  i32x8 g1;
  g1[0] = (2 << 16) | (1 << 20) | (5 << 22);         // data_size=4B, pad_en, interval=64DW, amount=1DW
  g1[1] = (int)(8u << 16);                           // tensor_dim0 = 8 (lo16 at [31:16])
  g1[2] = (int)(8u << 16);                           // tensor_dim0 hi=0 | tensor_dim1 = 8
  g1[3] = (int)(8u << 16);                           // tensor_dim1 hi=0 | tile_dim0 = 8
  g1[4] = (int)(8u | (64u << 16));                   // tile_dim1 = 8, tile_dim2 = 64
  g1[5] = 256;                                       // tensor_dim0_stride = 256 (lo32)
  g1[6] = 0;                                         // dim0_stride hi | dim1_stride lo16 = 0
  g1[7] = 1;                                         // tensor_dim1_stride = 65536 ([47:16] = 1)

  i32x4 g2; g2[0] = 64; g2[1] = 0; g2[2] = 0; g2[3] = 0;  // tensor_dim2 = 64
  i32x4 g3; g3[0] = 0;  g3[1] = 0; g3[2] = 0; g3[3] = 0;
  i32x8 g4 = {};                                          // unused extension arg: zero

  __builtin_amdgcn_tensor_load_to_lds(g0, g1, g2, g3, g4, 0);
}
#endif

// Stage XT[cs][jj][l] (stride 65) for both cells of the pair, dims
// j = jc*64 .. jc*64+63. Caller brackets with __syncthreads().
__device__ __forceinline__ void stage_chunk(const float* __restrict__ xb, float* XT,
                                            int gx, int gy0, int p, int jc,
                                            int tid, int wave) {
#if USE_TDM
  if (wave < 2) {
    const float* gsrc = xb + (size_t)(jc * 64) * (256 * 256) +
                        (size_t)(gx * 8) * 256 + (size_t)((gy0 + p * 2 + wave) * 8);
    tdm_load_cell_chunk(gsrc, XT + wave * (64 * 65));
  }
  __builtin_amdgcn_s_wait_tensorcnt(0);
#else
  for (int e = tid; e < 2 * 64 * 64; e += NTHREADS) {
    const int cs2 = e >> 12, r = e & 4095;
    const int jj = r >> 6, l = r & 63;
    const int ix = l >> 3, iy = l & 7;
    XT[cs2 * (64 * 65) + jj * 65 + l] =
        xb[(size_t)(jc * 64 + jj) * (256 * 256) +
           (size_t)(gx * 8 + ix) * 256 + (gy0 + p * 2 + cs2) * 8 + iy];
  }
  (void)wave;
#endif
}

__global__ __launch_bounds__(NTHREADS)
void grid_attn_kernel(const float* __restrict__ x,
                      const float* __restrict__ Wq, const float* __restrict__ bq,
                      const float* __restrict__ Wk, const float* __restrict__ bk,
                      const float* __restrict__ Wv, const float* __restrict__ bv,
                      const float* __restrict__ Wo, const float* __restrict__ bo,
                      float* __restrict__ out)
{
  extern __shared__ float lds[];
  const int tid  = threadIdx.x;
  const unsigned lane = tid & 31u;
  const int wave = tid >> 5;
  const int m16  = lane & 15;          // tile row (A) / col (B,D) for this lane
  const int kk   = (lane >> 4) << 1;   // k-pair base: 0 or 2 (ISA f32 A/B layout)
  const int mhf  = (lane >> 4) << 3;   // D row half: 0 or 8

  const int wg    = blockIdx.x;        // 512 blocks
  const int g0    = wg * CELLS;        // first global cell (b*1024 + gx*32 + gy)
  const int bIdx  = g0 >> 10;
  const int cell0 = g0 & 1023;
  const int gx    = cell0 >> 5;
  const int gy0   = cell0 & 31;        // multiple of 8 -> 8 cells share gx

  const float* xb = x + (size_t)bIdx * DM * 256 * 256;

  float* Q   = lds + OFF_Q;
  float* QP  = lds + OFF_QP;
  float* UP  = lds + OFF_U;
  float* CB  = lds + OFF_CB;
  float* XT  = lds + OFF_XT;
  float* AT  = lds + OFF_AT;
  float* S2  = lds + OFF_S2;
  float* CTX = QP;                     // QP dead after pair loop -> reuse

  // Zero padding rows 8..15 of Q (WMMA A-matrix M padding).
  for (int i = tid; i < 8 * 257; i += NTHREADS) Q[8 * 257 + i] = 0.f;

  // ---- Pass 0: avg-pool -> Q[c][j]. 8 gy-consecutive cells span 64
  // contiguous h-values per (j, row): fully coalesced 256B segments.
  {
    const size_t rowbase = (size_t)(gx * 8) * 256 + (size_t)(gy0 * 8);
    for (int j = wave; j < DM; j += 8) {
      const float* pj = xb + (size_t)j * (256 * 256) + rowbase;
      if (j + 8 < DM)   // prefetch next plane of the HBM stream
        __builtin_prefetch(pj + (size_t)8 * (256 * 256), 0, 1);
      float accA = 0.f, accB = 0.f;      // lane = column; cells 0-3 / 4-7
      #pragma unroll
      for (int ix = 0; ix < 8; ++ix) {
        accA += pj[ix * 256 + lane];
        accB += pj[ix * 256 + 32 + lane];
      }
      accA += __shfl_xor(accA, 1, 32); accA += __shfl_xor(accA, 2, 32); accA += __shfl_xor(accA, 4, 32);
      accB += __shfl_xor(accB, 1, 32); accB += __shfl_xor(accB, 2, 32); accB += __shfl_xor(accB, 4, 32);
      if ((lane & 7u) == 0u) {
        Q[(int)(lane >> 3) * 257 + j]       = accA * (1.f / 64.f);
        Q[(4 + (int)(lane >> 3)) * 257 + j] = accB * (1.f / 64.f);
      }
    }
  }
  __syncthreads();

  // ---- Pass 1: QP = Q @ Wq^T + bq  (M=16 padded cells, K=256, N=256)
  for (int t = wave; t < 16; t += 8) {
    v8f acc = {};
    const float* Wrow = Wq + (size_t)(t * 16 + m16) * DM;    // B[k][n] = Wq[n][k]
    for (int k0 = 0; k0 < DM; k0 += 4) {
      v2f a, b;
      a.x = Q[m16 * 257 + k0 + kk];  a.y = Q[m16 * 257 + k0 + kk + 1];
      b.x = Wrow[k0 + kk];           b.y = Wrow[k0 + kk + 1];
      acc = WMMA_F32(a, b, acc);
    }
    const float bias = bq[t * 16 + m16];
    #pragma unroll
    for (int i = 0; i < 8; ++i)
      QP[(mhf + i) * 257 + t * 16 + m16] = acc[i] + bias;
  }
  __syncthreads();

  // ---- score bias: cb[c][h] = qp_h . bk_h
  if (tid < 64) {
    const int c = tid >> 3, h = tid & 7;
    float s = 0.f;
    for (int i = 0; i < 32; ++i) s += QP[c * 257 + h * 32 + i] * bk[h * 32 + i];
    CB[c * 8 + h] = s;
  }

  const float scale = 0.17677669529663687f;  // 1/sqrt(dh=32)

  // ================= per cell-pair attention =================
  for (int p = 0; p < 4; ++p) {
    __syncthreads();

    // ---- U for this pair: per (head, j-tile), M rows = 2 cells (padded)
    for (int t = wave; t < 128; t += 8) {
      const int h = t >> 4, jt = t & 15;
      v8f acc = {};
      const float* Bb = Wk + (size_t)(h * 32) * DM + jt * 16;   // B[k][n]=Wk[h*32+k][jt*16+n]
      #pragma unroll
      for (int k0 = 0; k0 < 32; k0 += 4) {
        v2f a, b;
        const int row = (p * 2 + (m16 & 1)) * 257 + h * 32;
        a.x = QP[row + k0 + kk];        a.y = QP[row + k0 + kk + 1];
        b.x = Bb[(k0 + kk) * DM + m16]; b.y = Bb[(k0 + kk + 1) * DM + m16];
        acc = WMMA_F32(a, b, acc);
      }
      #pragma unroll
      for (int i = 0; i < 8; ++i) {
        const int m = mhf + i;
        if (m < 2) UP[m * (8 * 257) + h * 257 + jt * 16 + m16] = acc[i];
      }
    }
    __syncthreads();

    // ---- scores: 8 tiles = (cell-of-pair, token-tile); K=256 in 64-chunks
    const int cs = wave >> 2;            // which cell of the pair
    const int nt = wave & 3;             // token tile
    const int cl = p * 2 + cs;
    v8f sacc = {};
    for (int jc = 0; jc < 4; ++jc) {
      __syncthreads();                   // XT reuse fence
      stage_chunk(xb, XT, gx, gy0, p, jc, tid, wave);
      __syncthreads();
      const float* Ub = UP + cs * (8 * 257);
      const float* Xc = XT + cs * (64 * 65);
      for (int k0 = 0; k0 < 64; k0 += 4) {
        v2f a, b;
        a.x = Ub[(m16 & 7) * 257 + jc * 64 + k0 + kk];
        a.y = Ub[(m16 & 7) * 257 + jc * 64 + k0 + kk + 1];
        b.x = Xc[(k0 + kk) * 65 + nt * 16 + m16];
        b.y = Xc[(k0 + kk + 1) * 65 + nt * 16 + m16];
        sacc = WMMA_F32(a, b, sacc);
      }
    }
    #pragma unroll
    for (int i = 0; i < 8; ++i) {
      const int m = mhf + i;
      if (m < 8)
        AT[cs * (8 * 65) + m * 65 + nt * 16 + m16] = (sacc[i] + CB[cl * 8 + m]) * scale;
    }
    __syncthreads();

    // ---- softmax over 64 tokens, one (cell,head) row per thread
    if (tid < 16) {
      float* row = AT + (tid >> 3) * (8 * 65) + (tid & 7) * 65;
      float mx = row[0];
      for (int l = 1; l < 64; ++l) mx = fmaxf(mx, row[l]);
      float s = 0.f;
      for (int l = 0; l < 64; ++l) { const float e = expf(row[l] - mx); row[l] = e; s += e; }
      const float inv = 1.f / s;
      for (int l = 0; l < 64; ++l) row[l] *= inv;
    }
    __syncthreads();

    // ---- S2 = attn @ X : restage X chunks (L2-hot), K = 64 tokens
    for (int jc = 0; jc < 4; ++jc) {
      __syncthreads();
      stage_chunk(xb, XT, gx, gy0, p, jc, tid, wave);
      __syncthreads();
      const int cs3 = wave >> 2, jt = wave & 3;
      const int cl3 = p * 2 + cs3;
      v8f acc = {};
      const float* Ab = AT + cs3 * (8 * 65);
      const float* Xc = XT + cs3 * (64 * 65);
      #pragma unroll
      for (int k0 = 0; k0 < 64; k0 += 4) {
        v2f a, b;
        a.x = Ab[(m16 & 7) * 65 + k0 + kk];
        a.y = Ab[(m16 & 7) * 65 + k0 + kk + 1];
        b.x = Xc[(jt * 16 + m16) * 65 + k0 + kk];      // B[k=l][n=j] = XT[j][l]
        b.y = Xc[(jt * 16 + m16) * 65 + k0 + kk + 1];
        acc = WMMA_F32(a, b, acc);
      }
      #pragma unroll
      for (int i = 0; i < 8; ++i) {
        const int m = mhf + i;
        if (m < 8)
          S2[(cl3 * 8 + m) * 257 + jc * 64 + jt * 16 + m16] = acc[i];
      }
    }
  }
  __syncthreads();

  // ---- ctx[c, h*32+i] = Wv[h*32+i,:] . S2[c,h,:] + bv   (per-head GEMMs)
  for (int t = wave; t < 16; t += 8) {
    const int h = t >> 1, it = t & 1;
    v8f acc = {};
    const float* Wrow = Wv + (size_t)(h * 32 + it * 16 + m16) * DM;
    const float* Ab = S2 + h * 257;                     // row stride over cells = 8*257
    for (int k0 = 0; k0 < DM; k0 += 4) {
      v2f a, b;
      a.x = Ab[(m16 & 7) * (8 * 257) + k0 + kk];
      a.y = Ab[(m16 & 7) * (8 * 257) + k0 + kk + 1];
      b.x = Wrow[k0 + kk];  b.y = Wrow[k0 + kk + 1];
      acc = WMMA_F32(a, b, acc);
    }
    const float bias = bv[h * 32 + it * 16 + m16];
    #pragma unroll
    for (int i = 0; i < 8; ++i)
      CTX[(mhf + i) * 257 + h * 32 + it * 16 + m16] = acc[i] + bias;
  }
  __syncthreads();

  // ---- out = Q + CTX @ Wo^T + bo
  for (int t = wave; t < 16; t += 8) {
    v8f acc = {};
    const float* Wrow = Wo + (size_t)(t * 16 + m16) * DM;
    for (int k0 = 0; k0 < DM; k0 += 4) {
      v2f a, b;
      a.x = CTX[m16 * 257 + k0 + kk];  a.y = CTX[m16 * 257 + k0 + kk + 1];
      b.x = Wrow[k0 + kk];             b.y = Wrow[k0 + kk + 1];
      acc = WMMA_F32(a, b, acc);
    }
    const float bias = bo[t * 16 + m16];
    #pragma unroll
    for (int i = 0; i < 8; ++i) {
      const int m = mhf + i;
      if (m < 8) {
        const int nG = t * 16 + m16;
        out[(size_t)(g0 + m) * DM + nG] = acc[i] + bias + Q[m * 257 + nG];
      }
    }
  }
}

extern "C" void kernel_launch(void* const* d_in, const int* in_sizes, int n_in,
                              void* d_out, int out_size, void* d_ws, size_t ws_size,
                              hipStream_t stream) {
  const float* x  = (const float*)d_in[0];
  const float* Wq = (const float*)d_in[1];
  const float* bq = (const float*)d_in[2];
  const float* Wk = (const float*)d_in[3];
  const float* bk = (const float*)d_in[4];
  const float* Wv = (const float*)d_in[5];
  const float* bv = (const float*)d_in[6];
  const float* Wo = (const float*)d_in[7];
  const float* bo = (const float*)d_in[8];
  float* out = (float*)d_out;
  (void)in_sizes; (void)n_in; (void)out_size; (void)d_ws; (void)ws_size;

  const int total_cells = 4 * 32 * 32;             // B * gw * gh = 4096
  dim3 grid(total_cells / CELLS), block(NTHREADS); // 512 x 256
  const size_t shmem = (size_t)LDS_FLOATS * sizeof(float);
  hipLaunchKernelGGL(grid_attn_kernel, grid, block, shmem, stream,
                     x, Wq, bq, Wk, bk, Wv, bv, Wo, bo, out);
}